// FAGCN_15530601743022
// MI455X (gfx1250) — compile-verified
//
#include <hip/hip_runtime.h>
#include <hip/hip_bf16.h>
#include <math.h>

#define NN   50000
#define EE   1600000
#define HID  128
#define KITER 2
#define EPSV 0.3f
#define FIN  512
#define NCLS 40

typedef __attribute__((ext_vector_type(2))) float v2f;
typedef __attribute__((ext_vector_type(8))) float v8f;

// ---------------------------------------------------------------- utilities
__global__ void FAGCN_zero_kernel(float* __restrict__ p, long n) {
    long i = (long)blockIdx.x * blockDim.x + threadIdx.x;
    if (i < n) p[i] = 0.0f;
}

__global__ void FAGCN_deg_kernel(const int* __restrict__ row, float* __restrict__ deg) {
    int e = blockIdx.x * blockDim.x + threadIdx.x;
    if (e < EE) atomicAdd(&deg[row[e]], 1.0f);
}

__global__ void FAGCN_normdeg_kernel(float* __restrict__ deg) {
    int i = blockIdx.x * blockDim.x + threadIdx.x;
    if (i < NN) {
        float d = deg[i];
        d = d < 1.0f ? 1.0f : d;
        deg[i] = rsqrtf(d);   // clip(deg,1)^-0.5
    }
}

// agg := EPS * raw   (fused "EPS*raw + segment_sum" initialization)
__global__ void FAGCN_init_agg_kernel(const float* __restrict__ raw,
                                      float* __restrict__ agg) {
    long i = (long)blockIdx.x * blockDim.x + threadIdx.x;
    if (i < (long)NN * HID) agg[i] = EPSV * raw[i];
}

// -------------------------------------------------- GEMM1: raw = relu(x W1^T + b1)
// One wave per 16-row M tile; wave computes all 128 output cols (8 N-tiles).
// fp32 WMMA 16x16x4: A frag = float2/lane, B frag = float2/lane, C = v8f.
__global__ void __launch_bounds__(256)
FAGCN_gemm1_kernel(const float* __restrict__ x,    // N x 512
                   const float* __restrict__ w,    // 128 x 512 (row major)
                   const float* __restrict__ bias, // 128
                   float* __restrict__ raw) {      // N x 128
    const int lane  = threadIdx.x & 31;
    const int wave  = threadIdx.x >> 5;
    const int mt    = blockIdx.x * 8 + wave;
    const int m0    = mt * 16;
    if (m0 >= NN) return;                    // wave-uniform: EXEC stays all-ones

    const int rsel  = lane & 15;             // M (for A) / N (for B/C) within tile
    const int khalf = lane >> 4;             // 0: K=kk,kk+1   1: K=kk+2,kk+3

    int rowA = m0 + rsel; if (rowA > NN - 1) rowA = NN - 1;   // clamp tail loads
    const float* aBase = x + (size_t)rowA * FIN + 2 * khalf;
    const float* bBase = w + (size_t)rsel * FIN + 2 * khalf;

    v8f acc[8] = {};

    for (int kk = 0; kk < FIN; kk += 4) {
        v2f a = *(const v2f*)(aBase + kk);
#pragma unroll
        for (int t = 0; t < 8; ++t) {
            v2f bf = *(const v2f*)(bBase + (size_t)t * 16 * FIN + kk);
            acc[t] = __builtin_amdgcn_wmma_f32_16x16x4_f32(
                false, a, false, bf, (short)0, acc[t], false, false);
        }
    }

#pragma unroll
    for (int t = 0; t < 8; ++t) {
        const int n  = t * 16 + rsel;
        const float bv = bias[n];
#pragma unroll
        for (int r = 0; r < 8; ++r) {
            const int m = m0 + r + 8 * khalf;     // C/D layout: VGPR r -> M=r (+8 upper lanes)
            if (m < NN) {
                float v = acc[t][r] + bv;
                v = v > 0.0f ? v : 0.0f;          // relu
                raw[(size_t)m * HID + n] = v;
            }
        }
    }
}

// ------------------------------------ edge gate + normalized scatter (per k)
// One wave32 per edge; float4 per lane covers HID=128.
// agg must be pre-initialized to EPS*raw; after this kernel agg == next h.
__global__ void __launch_bounds__(256)
FAGCN_gate_scatter_kernel(const float* __restrict__ h,
                          const int*   __restrict__ row,
                          const int*   __restrict__ col,
                          const float* __restrict__ nd,
                          const float* __restrict__ gate_w,  // K x 256
                          const float* __restrict__ gate_b,  // K
                          int k,
                          float* __restrict__ agg) {
    const int lane = threadIdx.x & 31;
    const int wave = threadIdx.x >> 5;
    const long e   = (long)blockIdx.x * 8 + wave;
    if (e >= EE) return;                      // wave-uniform

    const int r = row[e];
    const int c = col[e];

    const float4 hr4 = ((const float4*)(h + (size_t)r * HID))[lane];
    const float4 hc4 = ((const float4*)(h + (size_t)c * HID))[lane];
    const float4 wr4 = ((const float4*)(gate_w + (size_t)k * 2 * HID))[lane];
    const float4 wc4 = ((const float4*)(gate_w + (size_t)k * 2 * HID + HID))[lane];

    float p = hr4.x * wr4.x + hr4.y * wr4.y + hr4.z * wr4.z + hr4.w * wr4.w
            + hc4.x * wc4.x + hc4.y * wc4.y + hc4.z * wc4.z + hc4.w * wc4.w;
#pragma unroll
    for (int off = 16; off; off >>= 1) p += __shfl_xor(p, off, 32);

    const float g   = tanhf(p + gate_b[k]);
    const float nrm = g * nd[r] * nd[c];

    float* dst = agg + (size_t)c * HID + lane * 4;
    atomicAdd(dst + 0, nrm * hr4.x);
    atomicAdd(dst + 1, nrm * hr4.y);
    atomicAdd(dst + 2, nrm * hr4.z);
    atomicAdd(dst + 3, nrm * hr4.w);
}

// ------------------------------------------- GEMM2 (N x 40) + log_softmax
__global__ void FAGCN_gemm2_kernel(const float* __restrict__ h,
                                   const float* __restrict__ w,   // 40 x 128
                                   const float* __restrict__ b,
                                   float* __restrict__ out) {
    long i = (long)blockIdx.x * blockDim.x + threadIdx.x;
    if (i >= (long)NN * NCLS) return;
    const int n = (int)(i / NCLS);
    const int c = (int)(i % NCLS);
    const float* hr = h + (size_t)n * HID;
    const float* wr = w + (size_t)c * HID;
    float s = 0.0f;
#pragma unroll 8
    for (int j = 0; j < HID; ++j) s = fmaf(hr[j], wr[j], s);
    out[i] = s + b[c];
}

__global__ void FAGCN_logsoftmax_kernel(float* __restrict__ out) {
    int n = blockIdx.x * blockDim.x + threadIdx.x;
    if (n >= NN) return;
    float* p = out + (size_t)n * NCLS;
    float m = p[0];
    for (int c = 1; c < NCLS; ++c) m = fmaxf(m, p[c]);
    float s = 0.0f;
    for (int c = 0; c < NCLS; ++c) s += expf(p[c] - m);
    const float l = m + logf(s);
    for (int c = 0; c < NCLS; ++c) p[c] = p[c] - l;
}

// ---------------------------------------------------------------- launcher
extern "C" void kernel_launch(void* const* d_in, const int* in_sizes, int n_in,
                              void* d_out, int out_size, void* d_ws, size_t ws_size,
                              hipStream_t stream) {
    const float* x      = (const float*)d_in[0];
    const int*   ei     = (const int*)  d_in[1];
    const float* t1_w   = (const float*)d_in[2];
    const float* t1_b   = (const float*)d_in[3];
    const float* gate_w = (const float*)d_in[4];
    const float* gate_b = (const float*)d_in[5];
    const float* t2_w   = (const float*)d_in[6];
    const float* t2_b   = (const float*)d_in[7];
    float* out = (float*)d_out;

    const int* row = ei;        // edge_index[0]
    const int* col = ei + EE;   // edge_index[1]

    float* ws   = (float*)d_ws;
    float* deg  = ws;                              // N floats, becomes normdeg in place
    float* raw  = ws + ((NN + 255) & ~255);        // N*HID
    float* agg0 = raw  + (size_t)NN * HID;         // N*HID (h after iter 0)
    float* agg1 = agg0 + (size_t)NN * HID;         // N*HID (h after iter 1)

    const long nh = (long)NN * HID;
    const int  nhBlocks = (int)((nh + 255) / 256);

    // degree -> norm_degree
    FAGCN_zero_kernel<<<(NN + 255) / 256, 256, 0, stream>>>(deg, (long)NN);
    FAGCN_deg_kernel<<<(EE + 255) / 256, 256, 0, stream>>>(row, deg);
    FAGCN_normdeg_kernel<<<(NN + 255) / 256, 256, 0, stream>>>(deg);

    // raw = relu(x @ t1_w^T + t1_b)
    const int mtiles = (NN + 15) / 16;
    FAGCN_gemm1_kernel<<<(mtiles + 7) / 8, 256, 0, stream>>>(x, t1_w, t1_b, raw);

    // iter 0: agg0 = EPS*raw + scatter(raw)   -> h = agg0
    FAGCN_init_agg_kernel<<<nhBlocks, 256, 0, stream>>>(raw, agg0);
    FAGCN_gate_scatter_kernel<<<(EE + 7) / 8, 256, 0, stream>>>(
        raw, row, col, deg, gate_w, gate_b, 0, agg0);

    // iter 1: agg1 = EPS*raw + scatter(agg0)  -> h = agg1
    FAGCN_init_agg_kernel<<<nhBlocks, 256, 0, stream>>>(raw, agg1);
    FAGCN_gate_scatter_kernel<<<(EE + 7) / 8, 256, 0, stream>>>(
        agg0, row, col, deg, gate_w, gate_b, 1, agg1);

    // logits + log_softmax
    FAGCN_gemm2_kernel<<<(int)(((long)NN * NCLS + 255) / 256), 256, 0, stream>>>(
        agg1, t2_w, t2_b, out);
    FAGCN_logsoftmax_kernel<<<(NN + 255) / 256, 256, 0, stream>>>(out);
}